// MyTripletLoss_11751030522609
// MI455X (gfx1250) — compile-verified
//
#include <hip/hip_runtime.h>
#include <float.h>
#include <math.h>

#define NN 4096
#define DD 512
#define NTILES (NN / 16)

typedef __attribute__((ext_vector_type(2))) float v2f;
typedef __attribute__((ext_vector_type(8))) float v8f;

// workspace layout (float units)
#define WS_SQA 0
#define WS_SQL (NN)
#define WS_DAP (2 * NN)
#define WS_DAN (3 * NN)
#define WS_JAP (4 * NN)
#define WS_JAN (5 * NN)
#define WS_ACC (6 * NN)

__global__ void init_accum_kernel(float* ws) {
    if (threadIdx.x < 4) ws[WS_ACC + threadIdx.x] = 0.0f;
}

__global__ __launch_bounds__(128) void row_sqnorm_kernel(const float* __restrict__ X,
                                                         float* __restrict__ sq) {
    __shared__ float red[128];
    const int row = blockIdx.x;
    const float* p = X + (size_t)row * DD;
    float s = 0.0f;
    for (int k = threadIdx.x; k < DD; k += 128) {
        float v = p[k];
        s += v * v;
    }
    red[threadIdx.x] = s;
    __syncthreads();
    for (int o = 64; o > 0; o >>= 1) {
        if (threadIdx.x < o) red[threadIdx.x] += red[threadIdx.x + o];
        __syncthreads();
    }
    if (threadIdx.x == 0) sq[row] = red[0];
}

// One workgroup per 32-row block (two 16-row A tiles staged in LDS).
// 8 waves split the 128 column-tile pairs; each wave computes a 2x2 grid of
// 16x16 Gram tiles per K pass with V_WMMA_F32_16X16X4_F32:
//   - each A fragment (LDS) feeds 2 WMMAs
//   - each B fragment (global) feeds 2 WMMAs
//   - 4 independent accumulator chains overlap load latency
// Tiles are converted to distances and hard-mined per-row (max positive,
// min negative) with first-occurrence argmax/argmin semantics.
__global__ __launch_bounds__(256) void gram_hard_mine_kernel(
    const float* __restrict__ X, const float* __restrict__ sq, const int* __restrict__ tgt,
    float* __restrict__ dap, float* __restrict__ dan, int* __restrict__ jap,
    int* __restrict__ jan) {
    __shared__ float ldsA[32 * DD];  // 64KB row block (rows rowBase..rowBase+31)
    __shared__ float s_pmax[8][32];
    __shared__ float s_nmin[8][32];
    __shared__ int s_pidx[8][32];
    __shared__ int s_nidx[8][32];

    const int rowBase = blockIdx.x * 32;
    const int tid = threadIdx.x;

    // Stage the 32xD row block into LDS (float4 cooperative copy).
    {
        const float4* src = (const float4*)(X + (size_t)rowBase * DD);
        float4* dst = (float4*)ldsA;
        for (int i = tid; i < 32 * DD / 4; i += 256) dst[i] = src[i];
    }
    __syncthreads();

    const int lane = tid & 31;
    const int wave = tid >> 5;
    const int m = lane & 15;
    const int half = lane >> 4;

    // Per-lane row constants: state slot s = rt*8 + r holds global row
    // rowBase + rt*16 + r + 8*half.
    float sqRow[16];
    int tRow[16];
#pragma unroll
    for (int rt = 0; rt < 2; ++rt) {
#pragma unroll
        for (int r = 0; r < 8; ++r) {
            const int gr = rowBase + rt * 16 + r + 8 * half;
            sqRow[rt * 8 + r] = sq[gr];
            tRow[rt * 8 + r] = tgt[gr];
        }
    }

    float pmax[16], nmin[16];
    int pidx[16], nidx[16];
#pragma unroll
    for (int s = 0; s < 16; ++s) {
        pmax[s] = -FLT_MAX;
        nmin[s] = FLT_MAX;
        pidx[s] = 0;
        nidx[s] = 0;
    }

    const float* Aptr0 = ldsA + m * DD + 2 * half;          // rows 0..15
    const float* Aptr1 = ldsA + (16 + m) * DD + 2 * half;   // rows 16..31

    for (int p = wave; p < NTILES / 2; p += 8) {
        const int col0 = (2 * p) * 16 + m;  // first tile's column for this lane
        const int col1 = col0 + 16;         // second tile's column
        const float* B0 = X + (size_t)col0 * DD + 2 * half;
        const float* B1 = X + (size_t)col1 * DD + 2 * half;

        // Warm caches for the next pair this wave will process (L2-resident).
        if (p + 8 < NTILES / 2) {
            __builtin_prefetch(B0 + 256 * (size_t)DD, 0, 0);
            __builtin_prefetch(B1 + 256 * (size_t)DD, 0, 0);
        }

        v8f acc00 = {0.f, 0.f, 0.f, 0.f, 0.f, 0.f, 0.f, 0.f};  // rows 0..15  x col0
        v8f acc01 = {0.f, 0.f, 0.f, 0.f, 0.f, 0.f, 0.f, 0.f};  // rows 0..15  x col1
        v8f acc10 = {0.f, 0.f, 0.f, 0.f, 0.f, 0.f, 0.f, 0.f};  // rows 16..31 x col0
        v8f acc11 = {0.f, 0.f, 0.f, 0.f, 0.f, 0.f, 0.f, 0.f};  // rows 16..31 x col1
#pragma unroll 4
        for (int k = 0; k < DD; k += 4) {
            v2f a0 = *(const v2f*)(Aptr0 + k);
            v2f a1 = *(const v2f*)(Aptr1 + k);
            v2f b0 = *(const v2f*)(B0 + k);
            v2f b1 = *(const v2f*)(B1 + k);
            acc00 = __builtin_amdgcn_wmma_f32_16x16x4_f32(false, a0, false, b0, (short)0,
                                                          acc00, false, false);
            acc01 = __builtin_amdgcn_wmma_f32_16x16x4_f32(false, a0, false, b1, (short)0,
                                                          acc01, false, false);
            acc10 = __builtin_amdgcn_wmma_f32_16x16x4_f32(false, a1, false, b0, (short)0,
                                                          acc10, false, false);
            acc11 = __builtin_amdgcn_wmma_f32_16x16x4_f32(false, a1, false, b1, (short)0,
                                                          acc11, false, false);
        }

        // Epilogue: per row tile, col0 before col1 (columns increase => strict
        // update keeps per-lane first occurrence).
        const float sqc0 = sq[col0];
        const float sqc1 = sq[col1];
        const int tc0 = tgt[col0];
        const int tc1 = tgt[col1];
#pragma unroll
        for (int rt = 0; rt < 2; ++rt) {
#pragma unroll
            for (int c = 0; c < 2; ++c) {
                const v8f acc = (rt == 0) ? ((c == 0) ? acc00 : acc01)
                                          : ((c == 0) ? acc10 : acc11);
                const int col = (c == 0) ? col0 : col1;
                const float sqc = (c == 0) ? sqc0 : sqc1;
                const int tc = (c == 0) ? tc0 : tc1;
#pragma unroll
                for (int r = 0; r < 8; ++r) {
                    const int s = rt * 8 + r;
                    const float d2 = sqRow[s] + sqc - 2.0f * acc[r];
                    const float dist = sqrtf(fmaxf(d2, 1e-12f));
                    const bool pos = (tRow[s] == tc);
                    const float vp = pos ? dist : -FLT_MAX;
                    const float vn = pos ? FLT_MAX : dist;
                    if (vp > pmax[s]) {
                        pmax[s] = vp;
                        pidx[s] = col;
                    }
                    if (vn < nmin[s]) {
                        nmin[s] = vn;
                        nidx[s] = col;
                    }
                }
            }
        }
    }

    // Cross-lane merge within each 16-lane half (tie-break: smaller index,
    // matching jnp.argmax/argmin first-occurrence).
#pragma unroll
    for (int off = 1; off < 16; off <<= 1) {
#pragma unroll
        for (int s = 0; s < 16; ++s) {
            float ov = __shfl_xor(pmax[s], off, 32);
            int oi = __shfl_xor(pidx[s], off, 32);
            if (ov > pmax[s] || (ov == pmax[s] && oi < pidx[s])) {
                pmax[s] = ov;
                pidx[s] = oi;
            }
            float nv = __shfl_xor(nmin[s], off, 32);
            int ni = __shfl_xor(nidx[s], off, 32);
            if (nv < nmin[s] || (nv == nmin[s] && ni < nidx[s])) {
                nmin[s] = nv;
                nidx[s] = ni;
            }
        }
    }

    if (m == 0) {
#pragma unroll
        for (int rt = 0; rt < 2; ++rt) {
#pragma unroll
            for (int r = 0; r < 8; ++r) {
                const int s = rt * 8 + r;
                const int lr = rt * 16 + r + 8 * half;  // row within block
                s_pmax[wave][lr] = pmax[s];
                s_pidx[wave][lr] = pidx[s];
                s_nmin[wave][lr] = nmin[s];
                s_nidx[wave][lr] = nidx[s];
            }
        }
    }
    __syncthreads();

    if (tid < 32) {
        float bp = -FLT_MAX;
        int bpi = 0x7fffffff;
        float bn = FLT_MAX;
        int bni = 0x7fffffff;
        for (int w = 0; w < 8; ++w) {
            float v = s_pmax[w][tid];
            int i = s_pidx[w][tid];
            if (v > bp || (v == bp && i < bpi)) {
                bp = v;
                bpi = i;
            }
            v = s_nmin[w][tid];
            i = s_nidx[w][tid];
            if (v < bn || (v == bn && i < bni)) {
                bn = v;
                bni = i;
            }
        }
        dap[rowBase + tid] = bp;
        jap[rowBase + tid] = bpi;
        dan[rowBase + tid] = bn;
        jan[rowBase + tid] = bni;
    }
}

// Per-row epilogue: recover buggy compressed indices, compute the two
// dist_local gathers via dot products, accumulate loss terms.
__global__ __launch_bounds__(256) void finalize_rows_kernel(
    const float* __restrict__ XL, const float* __restrict__ sqL, const int* __restrict__ tgt,
    const float* __restrict__ dap, const float* __restrict__ dan, const int* __restrict__ jap,
    const int* __restrict__ jan, float* __restrict__ accum) {
    __shared__ int ca_s[256];
    __shared__ int cn_s[256];
    __shared__ float d1_s[256];
    __shared__ float d2_s[256];
    __shared__ int s_idx[2];

    const int i = blockIdx.x;
    const int tid = threadIdx.x;
    const int ti = tgt[i];
    const int ja = jap[i];
    const int jn = jan[i];

    int ca = 0, cn = 0;
    for (int k = tid; k < NN; k += 256) {
        const int same = (tgt[k] == ti) ? 1 : 0;
        if (k <= ja) ca += same;
        if (k <= jn) cn += same;
    }
    ca_s[tid] = ca;
    cn_s[tid] = cn;
    __syncthreads();
    for (int o = 128; o > 0; o >>= 1) {
        if (tid < o) {
            ca_s[tid] += ca_s[tid + o];
            cn_s[tid] += cn_s[tid + o];
        }
        __syncthreads();
    }
    if (tid == 0) {
        // idx_ap = (#same-class <= j_ap) - 1 ; idx_an = (#diff-class <= j_an) - 1
        int ia = ca_s[0] - 1;
        int in_ = (jn + 1 - cn_s[0]) - 1;
        s_idx[0] = min(max(ia, 0), NN - 1);
        s_idx[1] = min(max(in_, 0), NN - 1);
    }
    __syncthreads();
    const int ia = s_idx[0];
    const int in_ = s_idx[1];

    const float* xi = XL + (size_t)i * DD;
    const float* xa = XL + (size_t)ia * DD;
    const float* xn = XL + (size_t)in_ * DD;
    float d1 = 0.f, d2 = 0.f;
    for (int k = tid; k < DD; k += 256) {
        const float v = xi[k];
        d1 += v * xa[k];
        d2 += v * xn[k];
    }
    d1_s[tid] = d1;
    d2_s[tid] = d2;
    __syncthreads();
    for (int o = 128; o > 0; o >>= 1) {
        if (tid < o) {
            d1_s[tid] += d1_s[tid + o];
            d2_s[tid] += d2_s[tid + o];
        }
        __syncthreads();
    }
    if (tid == 0) {
        const float sqi = sqL[i];
        const float ap_part = sqrtf(fmaxf(sqi + sqL[ia] - 2.f * d1_s[0], 1e-12f));
        const float an_part = sqrtf(fmaxf(sqi + sqL[in_] - 2.f * d2_s[0], 1e-12f));
        const float t_ori = fmaxf(dap[i] - dan[i], 0.f);
        const float t_part = fmaxf(ap_part - an_part, 0.f);
        const float prec = (dan[i] > dap[i]) ? 1.f : 0.f;
        atomicAdd(&accum[0], t_ori);
        atomicAdd(&accum[1], prec);
        atomicAdd(&accum[2], t_part);
    }
}

__global__ void write_out_kernel(const float* __restrict__ accum, float* __restrict__ out) {
    if (threadIdx.x == 0) {
        out[0] = accum[0] / (float)NN;  // loss_ori
        out[1] = accum[1] / (float)NN;  // prec_ori
        out[2] = accum[2] / (float)NN;  // loss_part
    }
}

extern "C" void kernel_launch(void* const* d_in, const int* in_sizes, int n_in, void* d_out,
                              int out_size, void* d_ws, size_t ws_size, hipStream_t stream) {
    const float* X = (const float*)d_in[0];   // inputs  [4096,512] f32
    const float* XL = (const float*)d_in[1];  // inputs_local
    const int* T = (const int*)d_in[2];       // targets

    float* ws = (float*)d_ws;
    float* sqA = ws + WS_SQA;
    float* sqL = ws + WS_SQL;
    float* dap = ws + WS_DAP;
    float* dan = ws + WS_DAN;
    int* jap = (int*)(ws + WS_JAP);
    int* jan = (int*)(ws + WS_JAN);
    float* accum = ws + WS_ACC;
    float* out = (float*)d_out;

    init_accum_kernel<<<1, 32, 0, stream>>>(ws);
    row_sqnorm_kernel<<<NN, 128, 0, stream>>>(X, sqA);
    row_sqnorm_kernel<<<NN, 128, 0, stream>>>(XL, sqL);
    gram_hard_mine_kernel<<<NN / 32, 256, 0, stream>>>(X, sqA, T, dap, dan, jap, jan);
    finalize_rows_kernel<<<NN, 256, 0, stream>>>(XL, sqL, T, dap, dan, jap, jan, accum);
    write_out_kernel<<<1, 32, 0, stream>>>(accum, out);
}